// DeBugModel_29351806501093
// MI455X (gfx1250) — compile-verified
//
#include <hip/hip_runtime.h>

typedef float v2f __attribute__((ext_vector_type(2)));
typedef float v8f __attribute__((ext_vector_type(8)));

#define NN 250000
#define NE 4000000
#define SLOPE 0.01f
#define BN_EPS 1e-5f
#define NW 4   // waves per node-MLP block (128 threads)

__device__ __forceinline__ float leakyf(float v) { return v > 0.f ? v : SLOPE * v; }

// ---------------------------------------------------------------------------
// 16x32 = (16x32 tile) @ (32x32 W, row-major W[k*32+n]) + bias, via
// V_WMMA_F32_16X16X4_F32 (full fp32: A 16x4 = v2f/lane, B 4x16 = v2f/lane,
// C/D 16x16 = v8f/lane per ISA layout tables). 2 column halves x 8 K-steps.
// ---------------------------------------------------------------------------
__device__ __forceinline__ void wmma_tile_gemm(const float* tin, const float* W,
                                               const float* bias, float* tout,
                                               int lane, bool act)
{
    const int half = lane >> 4;   // 0: lanes 0-15, 1: lanes 16-31
    const int l16  = lane & 15;   // A row (M) / B,C column (N) index
    #pragma unroll
    for (int n0 = 0; n0 < 32; n0 += 16) {
        float bb = bias[n0 + l16];
        v8f acc = { bb, bb, bb, bb, bb, bb, bb, bb };
        #pragma unroll
        for (int k0 = 0; k0 < 32; k0 += 4) {
            v2f a, bm;
            // A: lanes<16 hold K=k0+0,k0+1 ; lanes>=16 hold K=k0+2,k0+3 (M=l16)
            a.x  = tin[l16 * 32 + k0 + 2 * half + 0];
            a.y  = tin[l16 * 32 + k0 + 2 * half + 1];
            // B: same K split, N=n0+l16
            bm.x = W[(k0 + 2 * half + 0) * 32 + n0 + l16];
            bm.y = W[(k0 + 2 * half + 1) * 32 + n0 + l16];
            acc = __builtin_amdgcn_wmma_f32_16x16x4_f32(false, a, false, bm,
                                                        (short)0, acc, false, false);
        }
        #pragma unroll
        for (int r = 0; r < 8; ++r) {
            float v = acc[r];
            if (act) v = leakyf(v);
            // C: VGPR r -> M = r (lanes<16) / r+8 (lanes>=16), N = n0+l16
            tout[(half * 8 + r) * 32 + n0 + l16] = v;
        }
    }
    __builtin_amdgcn_wave_barrier();
}

// ---------------------------------------------------------------------------
// Edge message kernels:  m = relu(h[src] + edge_attr*We + be) ; agg[dst] += m
// h and agg are L2-resident (32 MB each) so random access mostly hits L2.
// ---------------------------------------------------------------------------
__global__ void edge_msg1(const int* __restrict__ src, const int* __restrict__ dst,
                          const float* __restrict__ x, const float* __restrict__ ea,
                          const float* __restrict__ We, const float* __restrict__ be,
                          float* __restrict__ agg)
{
    int j = blockIdx.x * blockDim.x + threadIdx.x;
    if (j >= NE) return;
    float m = x[src[j]] + ea[j] * We[0] + be[0];
    m = m > 0.f ? m : 0.f;
    atomicAdd(&agg[dst[j]], m);
}

__global__ void edge_msg32(const int* __restrict__ src, const int* __restrict__ dst,
                           const float* __restrict__ h, const float* __restrict__ ea,
                           const float* __restrict__ We, const float* __restrict__ be,
                           float* __restrict__ agg)
{
    long long t = (long long)blockIdx.x * blockDim.x + threadIdx.x;
    int j = (int)(t >> 5);
    int c = (int)(t & 31);
    if (j >= NE) return;
    float m = h[(long long)src[j] * 32 + c] + ea[j] * We[c] + be[c];
    m = m > 0.f ? m : 0.f;
    atomicAdd(&agg[(long long)dst[j] * 32 + c], m);
}

// ---------------------------------------------------------------------------
// Node MLP kernels (WMMA). One wave = one 16-node tile.
// ---------------------------------------------------------------------------
__global__ void __launch_bounds__(128)
node_mlp1(const float* __restrict__ x, const float* __restrict__ agg,
          const float* __restrict__ eps,
          const float* __restrict__ W1row, const float* __restrict__ b1,
          const float* __restrict__ W2, const float* __restrict__ b2,
          const float* __restrict__ W3, const float* __restrict__ b3,
          float* __restrict__ out, int numTiles)
{
    __shared__ float sW2[1024], sW3[1024];
    __shared__ float sw1[32], sb1[32], sb2[32], sb3[32];
    __shared__ float tiles[NW][2][512];
    int tid = threadIdx.x;
    for (int i = tid; i < 1024; i += blockDim.x) { sW2[i] = W2[i]; sW3[i] = W3[i]; }
    if (tid < 32) { sw1[tid] = W1row[tid]; sb1[tid] = b1[tid]; sb2[tid] = b2[tid]; sb3[tid] = b3[tid]; }
    __syncthreads();
    float ep1 = 1.0f + eps[0];
    int wave = tid >> 5, lane = tid & 31;
    int tile = blockIdx.x * NW + wave;
    if (tile >= numTiles) return;
    int base = tile * 16;
    float* t0 = tiles[wave][0];
    float* t1 = tiles[wave][1];
    #pragma unroll
    for (int i = 0; i < 16; ++i) {
        float v = ep1 * x[base + i] + agg[base + i];          // scalar input
        t0[i * 32 + lane] = leakyf(v * sw1[lane] + sb1[lane]); // 1->32 expansion
    }
    __builtin_amdgcn_wave_barrier();
    wmma_tile_gemm(t0, sW2, sb2, t1, lane, true);
    wmma_tile_gemm(t1, sW3, sb3, t0, lane, false);
    #pragma unroll
    for (int i = 0; i < 16; ++i)
        out[(base + i) * 32 + lane] = t0[i * 32 + lane];
}

__global__ void __launch_bounds__(128)
node_mlp32(const float* __restrict__ h, const float* __restrict__ agg,
           const float* __restrict__ eps,
           const float* __restrict__ W1, const float* __restrict__ b1,
           const float* __restrict__ W2, const float* __restrict__ b2,
           const float* __restrict__ W3, const float* __restrict__ b3,
           float* __restrict__ out, int numTiles)
{
    __shared__ float sW1[1024], sW2[1024], sW3[1024];
    __shared__ float sb1[32], sb2[32], sb3[32];
    __shared__ float tiles[NW][2][512];
    int tid = threadIdx.x;
    for (int i = tid; i < 1024; i += blockDim.x) { sW1[i] = W1[i]; sW2[i] = W2[i]; sW3[i] = W3[i]; }
    if (tid < 32) { sb1[tid] = b1[tid]; sb2[tid] = b2[tid]; sb3[tid] = b3[tid]; }
    __syncthreads();
    float ep1 = 1.0f + eps[0];
    int wave = tid >> 5, lane = tid & 31;
    int tile = blockIdx.x * NW + wave;
    if (tile >= numTiles) return;
    int base = tile * 16;
    float* t0 = tiles[wave][0];
    float* t1 = tiles[wave][1];
    #pragma unroll
    for (int i = 0; i < 16; ++i) {
        long long n = base + i;
        t0[i * 32 + lane] = ep1 * h[n * 32 + lane] + agg[n * 32 + lane];
    }
    __builtin_amdgcn_wave_barrier();
    wmma_tile_gemm(t0, sW1, sb1, t1, lane, true);
    wmma_tile_gemm(t1, sW2, sb2, t0, lane, true);
    wmma_tile_gemm(t0, sW3, sb3, t1, lane, false);
    #pragma unroll
    for (int i = 0; i < 16; ++i)
        out[(long long)(base + i) * 32 + lane] = t1[i * 32 + lane];
}

__global__ void __launch_bounds__(128)
node_mlp3(const float* __restrict__ h, const float* __restrict__ agg,
          const float* __restrict__ eps,
          const float* __restrict__ W1, const float* __restrict__ b1,
          const float* __restrict__ W2, const float* __restrict__ b2,
          const float* __restrict__ W3col, const float* __restrict__ b3,
          float* __restrict__ out, int numTiles)
{
    __shared__ float sW1[1024], sW2[1024];
    __shared__ float sb1[32], sb2[32], sw3[32];
    __shared__ float tiles[NW][2][512];
    int tid = threadIdx.x;
    for (int i = tid; i < 1024; i += blockDim.x) { sW1[i] = W1[i]; sW2[i] = W2[i]; }
    if (tid < 32) { sb1[tid] = b1[tid]; sb2[tid] = b2[tid]; sw3[tid] = W3col[tid]; }
    __syncthreads();
    float ep1 = 1.0f + eps[0];
    int wave = tid >> 5, lane = tid & 31;
    int tile = blockIdx.x * NW + wave;
    if (tile >= numTiles) return;
    int base = tile * 16;
    float* t0 = tiles[wave][0];
    float* t1 = tiles[wave][1];
    #pragma unroll
    for (int i = 0; i < 16; ++i) {
        long long n = base + i;
        t0[i * 32 + lane] = ep1 * h[n * 32 + lane] + agg[n * 32 + lane];
    }
    __builtin_amdgcn_wave_barrier();
    wmma_tile_gemm(t0, sW1, sb1, t1, lane, true);
    wmma_tile_gemm(t1, sW2, sb2, t0, lane, true);
    if (lane < 16) {                       // 32 -> 1 projection (after WMMA)
        float acc = b3[0];
        #pragma unroll
        for (int c = 0; c < 32; ++c) acc += t0[lane * 32 + c] * sw3[c];
        out[base + lane] = acc;
    }
}

// ---------------------------------------------------------------------------
// BatchNorm (full-batch, training-style mean/var over N) kernels.
// ---------------------------------------------------------------------------
__global__ void bn_reduce32(const float* __restrict__ h, float* __restrict__ stats)
{
    __shared__ float ssum[32], ssq[32];
    int tid = threadIdx.x;
    if (tid < 32) { ssum[tid] = 0.f; ssq[tid] = 0.f; }
    __syncthreads();
    int c = tid & 31;
    int rpb = blockDim.x >> 5;
    float s = 0.f, q = 0.f;
    for (int n = blockIdx.x * rpb + (tid >> 5); n < NN; n += gridDim.x * rpb) {
        float v = h[(long long)n * 32 + c];
        s += v; q += v * v;
    }
    atomicAdd(&ssum[c], s); atomicAdd(&ssq[c], q);
    __syncthreads();
    if (tid < 32) { atomicAdd(&stats[tid], ssum[tid]); atomicAdd(&stats[32 + tid], ssq[tid]); }
}

__global__ void bn_finalize32(const float* __restrict__ g, const float* __restrict__ b,
                              float* __restrict__ stats)
{
    int c = threadIdx.x;
    if (c >= 32) return;
    float mean = stats[c] * (1.0f / (float)NN);
    float var  = stats[32 + c] * (1.0f / (float)NN) - mean * mean;
    float inv  = rsqrtf(var + BN_EPS);
    float sc   = inv * g[c];
    stats[64 + c] = sc;
    stats[96 + c] = b[c] - mean * sc;
}

__global__ void bn_apply_relu32(float* __restrict__ h, const float* __restrict__ stats)
{
    int t = blockIdx.x * blockDim.x + threadIdx.x;
    if (t >= NN * 32) return;
    int c = t & 31;
    float v = h[t] * stats[64 + c] + stats[96 + c];
    h[t] = v > 0.f ? v : 0.f;
}

__global__ void bn_reduce1(const float* __restrict__ v, float* __restrict__ stats)
{
    __shared__ float ss[2];
    int tid = threadIdx.x;
    if (tid == 0) { ss[0] = 0.f; ss[1] = 0.f; }
    __syncthreads();
    float s = 0.f, q = 0.f;
    for (int n = blockIdx.x * blockDim.x + tid; n < NN; n += gridDim.x * blockDim.x) {
        float x = v[n]; s += x; q += x * x;
    }
    atomicAdd(&ss[0], s); atomicAdd(&ss[1], q);
    __syncthreads();
    if (tid == 0) { atomicAdd(&stats[0], ss[0]); atomicAdd(&stats[1], ss[1]); }
}

__global__ void bn_finalize1(const float* __restrict__ g, const float* __restrict__ b,
                             float* __restrict__ stats)
{
    if (threadIdx.x != 0) return;
    float mean = stats[0] * (1.0f / (float)NN);
    float var  = stats[1] * (1.0f / (float)NN) - mean * mean;
    float inv  = rsqrtf(var + BN_EPS);
    float sc   = inv * g[0];
    stats[2] = sc;
    stats[3] = b[0] - mean * sc;
}

__global__ void bn_apply1(const float* __restrict__ v, const float* __restrict__ stats,
                          float* __restrict__ out)
{
    int n = blockIdx.x * blockDim.x + threadIdx.x;
    if (n >= NN) return;
    out[n] = v[n] * stats[2] + stats[3];
}

// ---------------------------------------------------------------------------
// Host launcher. Input order = flattened setup_inputs dict:
// 0:x 1:edge_index 2:edge_attr
// conv1: 3:eps 4:We 5:be 6:W1 7:b1 8:W2 9:b2 10:W3 11:b3
// conv2: 12..20   conv3: 21..29
// 30:bn1.g 31:bn1.b 32:bn2.g 33:bn2.b 34:bn3.g 35:bn3.b
// ---------------------------------------------------------------------------
extern "C" void kernel_launch(void* const* d_in, const int* in_sizes, int n_in,
                              void* d_out, int out_size, void* d_ws, size_t ws_size,
                              hipStream_t stream)
{
    (void)in_sizes; (void)n_in; (void)out_size; (void)ws_size;
    const float* x   = (const float*)d_in[0];
    const int*   ei  = (const int*)d_in[1];
    const int*   src = ei;
    const int*   dst = ei + NE;
    const float* ea  = (const float*)d_in[2];

    const float* c1[9]; const float* c2[9]; const float* c3[9];
    for (int i = 0; i < 9; ++i) { c1[i] = (const float*)d_in[3 + i];
                                  c2[i] = (const float*)d_in[12 + i];
                                  c3[i] = (const float*)d_in[21 + i]; }
    // index within conv: 0:eps 1:We 2:be 3:W1 4:b1 5:W2 6:b2 7:W3 8:b3
    const float* bn1g = (const float*)d_in[30]; const float* bn1b = (const float*)d_in[31];
    const float* bn2g = (const float*)d_in[32]; const float* bn2b = (const float*)d_in[33];
    const float* bn3g = (const float*)d_in[34]; const float* bn3b = (const float*)d_in[35];

    float* agg   = (float*)d_ws;                    // N*32
    float* hA    = agg + (size_t)NN * 32;           // N*32
    float* hB    = hA  + (size_t)NN * 32;           // N*32
    float* stats = hB  + (size_t)NN * 32;           // 128

    const int numTiles = NN / 16;                   // 15625 (exact)
    dim3 nodeGrid((numTiles + NW - 1) / NW);
    dim3 edgeGrid1((NE + 255) / 256);
    dim3 edgeGrid32((NE * 32LL) / 256);
    dim3 nwGrid(((long long)NN * 32 + 255) / 256);

    // ---- conv1 (d_in = 1) ----
    hipMemsetAsync(agg, 0, (size_t)NN * sizeof(float), stream);
    edge_msg1<<<edgeGrid1, 256, 0, stream>>>(src, dst, x, ea, c1[1], c1[2], agg);
    node_mlp1<<<nodeGrid, 128, 0, stream>>>(x, agg, c1[0], c1[3], c1[4], c1[5], c1[6],
                                            c1[7], c1[8], hA, numTiles);
    hipMemsetAsync(stats, 0, 128 * sizeof(float), stream);
    bn_reduce32<<<1024, 256, 0, stream>>>(hA, stats);
    bn_finalize32<<<1, 32, 0, stream>>>(bn1g, bn1b, stats);
    bn_apply_relu32<<<nwGrid, 256, 0, stream>>>(hA, stats);

    // ---- conv2 (d_in = 32) ----
    hipMemsetAsync(agg, 0, (size_t)NN * 32 * sizeof(float), stream);
    edge_msg32<<<edgeGrid32, 256, 0, stream>>>(src, dst, hA, ea, c2[1], c2[2], agg);
    node_mlp32<<<nodeGrid, 128, 0, stream>>>(hA, agg, c2[0], c2[3], c2[4], c2[5], c2[6],
                                             c2[7], c2[8], hB, numTiles);
    hipMemsetAsync(stats, 0, 128 * sizeof(float), stream);
    bn_reduce32<<<1024, 256, 0, stream>>>(hB, stats);
    bn_finalize32<<<1, 32, 0, stream>>>(bn2g, bn2b, stats);
    bn_apply_relu32<<<nwGrid, 256, 0, stream>>>(hB, stats);

    // ---- conv3 (d_out = 1) ----
    hipMemsetAsync(agg, 0, (size_t)NN * 32 * sizeof(float), stream);
    edge_msg32<<<edgeGrid32, 256, 0, stream>>>(src, dst, hB, ea, c3[1], c3[2], agg);
    node_mlp3<<<nodeGrid, 128, 0, stream>>>(hB, agg, c3[0], c3[3], c3[4], c3[5], c3[6],
                                            c3[7], c3[8], hA, numTiles);  // hA[:N] scalar
    hipMemsetAsync(stats, 0, 8 * sizeof(float), stream);
    bn_reduce1<<<512, 256, 0, stream>>>(hA, stats);
    bn_finalize1<<<1, 32, 0, stream>>>(bn3g, bn3b, stats);
    bn_apply1<<<(NN + 255) / 256, 256, 0, stream>>>(hA, stats, (float*)d_out);
}